// MyNet_51135880626479
// MI455X (gfx1250) — compile-verified
//
#include <hip/hip_runtime.h>
#include <cstdint>

// ---------------------------------------------------------------------------
// Fused block-diagonal MLP head for MI455X (gfx1250).
//   y_b = sigmoid( BNf( sum_f relu(xo_{b,f}) * c_f + K ) )
// c_f and K fold BN + fc dot into per-feature coefficients.
// Main pass: WMMA f32 16x16x4 per 6x6 node block (A=W, B=x^T, C=bias),
// x streamed through LDS with global_load_async_to_lds_b64 double buffering.
// ---------------------------------------------------------------------------

namespace {
constexpr int   kNodes  = 17185;
constexpr int   kD      = 6;
constexpr int   kFeats  = kNodes * kD;               // 103110
constexpr int   kBatch  = 256;
constexpr float kEps    = 1e-5f;

constexpr int   kNC     = 128;                       // nodes per LDS chunk
constexpr int   kNChunk = (kNodes + kNC - 1) / kNC;  // 135
constexpr int   kPitchF = kNC * kD + 4;              // 772 floats/row (pad -> bank-conflict-free reads)
constexpr int   kPitchB = kPitchF * 4;               // bytes/row
constexpr int   kGroups = 24;                        // chunk groups (grid.x of main kernel)
constexpr int   kKB     = (kFeats + 255) / 256;      // 403 blocks for coef/K pass
}

typedef __attribute__((ext_vector_type(2))) float v2f;
typedef __attribute__((ext_vector_type(8))) float v8f;

// WMMA wrapper: builtin when the (device) compiler declares it, inline asm
// fallback otherwise (host pass parses the asm branch, never codegens it).
__device__ __forceinline__ v8f wmma_f32_16x16x4(v2f a, v2f b, v8f c) {
#if __has_builtin(__builtin_amdgcn_wmma_f32_16x16x4_f32)
  return __builtin_amdgcn_wmma_f32_16x16x4_f32(
      /*neg_a=*/false, a, /*neg_b=*/false, b,
      /*c_mod=*/(short)0, c, /*reuse_a=*/false, /*reuse_b=*/false);
#else
  v8f d;
  asm volatile("v_wmma_f32_16x16x4_f32 %0, %1, %2, %3"
               : "=v"(d) : "v"(a), "v"(b), "v"(c));
  return d;
#endif
}

// CDNA5 async global->LDS copy (per-lane b64), tracked by ASYNCcnt.
__device__ __forceinline__ void asyncLoadB64(unsigned ldsByteAddr, const float* gp) {
  asm volatile("global_load_async_to_lds_b64 %0, %1, off"
               :: "v"(ldsByteAddr), "v"(gp) : "memory");
}
__device__ __forceinline__ void waitAsync0() {
  asm volatile("s_wait_asynccnt 0" ::: "memory");
}

// ---------------------------------------------------------------------------
// Kernel 1: build packed per-node record bc[n*12] = {bias[0..5], c[0..5]}
// with c = rsqrt(var+eps)*gamma*fcw, and partial sums of
// t = (beta - mean*rsqrt*gamma)*fcw (for constant K).
// ---------------------------------------------------------------------------
__global__ __launch_bounds__(256) void coef_kernel(
    const float* __restrict__ gamma, const float* __restrict__ beta,
    const float* __restrict__ mean,  const float* __restrict__ var,
    const float* __restrict__ fcw,   const float* __restrict__ bb,
    float* __restrict__ bc,          float* __restrict__ kpart)
{
  __shared__ float sm[256];
  const int tid = threadIdx.x;
  const int f   = blockIdx.x * 256 + tid;
  float t = 0.f;
  if (f < kFeats) {
    const float s = rsqrtf(var[f] + kEps) * gamma[f];
    const float w = fcw[f];
    const int   n = f / kD;
    const int   o = f - n * kD;
    bc[n * 12 + o]      = bb[f];   // bias
    bc[n * 12 + 6 + o]  = s * w;   // folded coefficient
    t = (beta[f] - mean[f] * s) * w;
  }
  sm[tid] = t;
  __syncthreads();
  for (int o = 128; o > 0; o >>= 1) {
    if (tid < o) sm[tid] += sm[tid + o];
    __syncthreads();
  }
  if (tid == 0) kpart[blockIdx.x] = sm[0];
}

// Kernel 2: K = sum(kpart) + fc_b
__global__ __launch_bounds__(256) void kred_kernel(
    const float* __restrict__ kpart, const float* __restrict__ fcb,
    float* __restrict__ ktot)
{
  __shared__ float sm[256];
  const int tid = threadIdx.x;
  float s = 0.f;
  for (int i = tid; i < kKB; i += 256) s += kpart[i];
  sm[tid] = s;
  __syncthreads();
  for (int o = 128; o > 0; o >>= 1) {
    if (tid < o) sm[tid] += sm[tid + o];
    __syncthreads();
  }
  if (tid == 0) ktot[0] = sm[0] + fcb[0];
}

// ---------------------------------------------------------------------------
// Main fused pass. Grid: (kGroups, kBatch/16). Block: 256 (= 8 wave32).
// Each wave: 16 nodes/chunk x 16 batch columns via two chained
// V_WMMA_F32_16X16X4_F32 (K = 4 + 2, padded with zeros).
// ---------------------------------------------------------------------------
__device__ __forceinline__ void prefetch_tile(const float* __restrict__ data,
                                              int b0, int n0, unsigned ldsBase,
                                              int wv, int lane)
{
  const int e0 = n0 * kD;  // first feature column of this chunk
#pragma unroll
  for (int rr = 0; rr < 2; ++rr) {
    const int row = wv * 2 + rr;                     // batch row within tile
    const float* gp = data + (size_t)(b0 + row) * kFeats + e0 + lane * 2;
    const unsigned lb = ldsBase + (unsigned)(row * kPitchB + lane * 8);
    if (e0 + kNC * kD <= kFeats) {                   // full chunk: unguarded
#pragma unroll
      for (int j = 0; j < 12; ++j) asyncLoadB64(lb + j * 256, gp + j * 64);
    } else {                                         // tail chunk: per-lane guard
#pragma unroll
      for (int j = 0; j < 12; ++j)
        if (e0 + lane * 2 + j * 64 < kFeats) asyncLoadB64(lb + j * 256, gp + j * 64);
    }
  }
}

__global__ __launch_bounds__(256) void fused_main(
    const float* __restrict__ data, const float* __restrict__ wb,
    const float* __restrict__ bc,   float* __restrict__ partials)
{
  __shared__ __align__(16) float xs[2][16 * kPitchF];  // double-buffered x tiles
  __shared__ float red[128];

  const int tid  = threadIdx.x;
  const int wv   = tid >> 5;
  const int lane = tid & 31;
  const int hi   = lane >> 4;   // lane half selects K-group (A/B) / M-row half (D)
  const int lr   = lane & 15;   // M row (A) / batch column (B,D)
  const int g    = blockIdx.x;
  const int b0   = blockIdx.y * 16;

  const unsigned base0 = (unsigned)(uintptr_t)&xs[0][0];
  const unsigned base1 = (unsigned)(uintptr_t)&xs[1][0];

  prefetch_tile(data, b0, g * kNC, base0, wv, lane);

  float acc = 0.f;

  // Per-node body: two chained WMMAs + relu/scale accumulate.
  auto node_body = [&](int n, int nl, const float* xrow) {
    // A = W[n] : lanes 0-15 rows o, K={0,1}/{4,5}; lanes 16-31 K={2,3}/0
    const float* wp = wb + (size_t)n * 36;
    v2f a0 = {0.f, 0.f}, a1 = {0.f, 0.f};
    if (lr < kD) {
      a0 = *(const v2f*)(wp + lr * 6 + hi * 2);
      if (!hi) a1 = *(const v2f*)(wp + lr * 6 + 4);
    }
    // B = x^T from LDS: lane n<16 -> {i0,i1}/{i4,i5}; lane>=16 -> {i2,i3}/0
    const float* xr = xrow + nl * 6;
    v2f bm0 = *(const v2f*)(xr + hi * 2);
    v2f bm1 = {0.f, 0.f};
    if (!hi) bm1 = *(const v2f*)(xr + 4);
    // Packed per-node record: [bias0..5, c0..5]
    const float* pc = bc + (size_t)n * 12;
    v8f cc = {pc[0], pc[1], pc[2], pc[3], pc[4], pc[5], 0.f, 0.f};
    v8f d = wmma_f32_16x16x4(a0, bm0, cc);
    d = wmma_f32_16x16x4(a1, bm1, d);
    acc = fmaf(fmaxf(d[0], 0.f), pc[6],  acc);
    acc = fmaf(fmaxf(d[1], 0.f), pc[7],  acc);
    acc = fmaf(fmaxf(d[2], 0.f), pc[8],  acc);
    acc = fmaf(fmaxf(d[3], 0.f), pc[9],  acc);
    acc = fmaf(fmaxf(d[4], 0.f), pc[10], acc);
    acc = fmaf(fmaxf(d[5], 0.f), pc[11], acc);
  };

  int stage = 0;
  for (int k = g; k < kNChunk; k += kGroups) {
    waitAsync0();          // this wave's tile writes landed
    __syncthreads();       // whole tile resident; previous buffer free (WAR)

    const int nk = k + kGroups;
    if (nk < kNChunk)      // overlap next tile's DMA with compute
      prefetch_tile(data, b0, nk * kNC, (stage ? base0 : base1), wv, lane);

    const float* xrow = &xs[stage][0] + lr * kPitchF;
    const int nbase = k * kNC + wv * 16;
    if (nbase + 16 <= kNodes) {           // full sub-chunk: no per-node guard
#pragma unroll 4
      for (int j = 0; j < 16; ++j) node_body(nbase + j, wv * 16 + j, xrow);
    } else {                              // tail: wave-uniform guard per node
#pragma unroll 4
      for (int j = 0; j < 16; ++j)
        if (nbase + j < kNodes) node_body(nbase + j, wv * 16 + j, xrow);
    }
    stage ^= 1;
  }

  // block reduce: 8 waves' per-batch partials (lanes 0-15 hold batch b0+lr)
  if (hi == 0) red[wv * 16 + lr] = acc;
  __syncthreads();
  if (tid < 16) {
    float s = 0.f;
#pragma unroll
    for (int w2 = 0; w2 < 8; ++w2) s += red[w2 * 16 + tid];
    partials[(size_t)(b0 + tid) * kGroups + g] = s;
  }
}

// ---------------------------------------------------------------------------
// Kernel 4: final per-batch reduce + scalar BN + sigmoid
// ---------------------------------------------------------------------------
__global__ __launch_bounds__(256) void final_kernel(
    const float* __restrict__ partials, const float* __restrict__ ktot,
    const float* __restrict__ bfg, const float* __restrict__ bfb,
    const float* __restrict__ bfm, const float* __restrict__ bfv,
    float* __restrict__ out)
{
  const int b = threadIdx.x;  // 256 threads == batch
  float y = ktot[0];
#pragma unroll 4
  for (int g = 0; g < kGroups; ++g) y += partials[(size_t)b * kGroups + g];
  const float z = (y - bfm[0]) * rsqrtf(bfv[0] + kEps) * bfg[0] + bfb[0];
  out[b] = 1.0f / (1.0f + expf(-z));
}

// ---------------------------------------------------------------------------
extern "C" void kernel_launch(void* const* d_in, const int* in_sizes, int n_in,
                              void* d_out, int out_size, void* d_ws, size_t ws_size,
                              hipStream_t stream)
{
  (void)in_sizes; (void)n_in; (void)out_size; (void)ws_size;
  const float* data = (const float*)d_in[0];
  const float* wb   = (const float*)d_in[1];
  const float* bb   = (const float*)d_in[2];
  const float* bng  = (const float*)d_in[3];
  const float* bnb  = (const float*)d_in[4];
  const float* bnm  = (const float*)d_in[5];
  const float* bnv  = (const float*)d_in[6];
  const float* fcw  = (const float*)d_in[7];
  const float* fcb  = (const float*)d_in[8];
  const float* bfg  = (const float*)d_in[9];
  const float* bfb  = (const float*)d_in[10];
  const float* bfm  = (const float*)d_in[11];
  const float* bfv  = (const float*)d_in[12];

  float* ws       = (float*)d_ws;
  float* bc       = ws;                    // kNodes * 12 (packed bias+coef)
  float* kpart    = bc + (size_t)kNodes * 12;
  float* ktot     = kpart + kKB;           // 1
  float* partials = ktot + 1;              // kBatch * kGroups

  coef_kernel<<<kKB, 256, 0, stream>>>(bng, bnb, bnm, bnv, fcw, bb, bc, kpart);
  kred_kernel<<<1, 256, 0, stream>>>(kpart, fcb, ktot);
  fused_main<<<dim3(kGroups, kBatch / 16), 256, 0, stream>>>(data, wb, bc, partials);
  final_kernel<<<1, 256, 0, stream>>>(partials, ktot, bfg, bfb, bfm, bfv, (float*)d_out);
}